// GNNModel_30520037605641
// MI455X (gfx1250) — compile-verified
//
#include <hip/hip_runtime.h>
#include <hip/hip_bf16.h>

typedef __attribute__((ext_vector_type(2))) float v2f;
typedef __attribute__((ext_vector_type(8))) float v8f;

#define FDIM 128

// ---------------- utility kernels ----------------

__global__ void zero_kernel(float* __restrict__ p, int n) {
  int i = blockIdx.x * blockDim.x + threadIdx.x;
  if (i < n) p[i] = 0.0f;
}

__global__ void deg_kernel(const int* __restrict__ dst, float* __restrict__ deg, int E) {
  int e = blockIdx.x * blockDim.x + threadIdx.x;
  if (e < E) atomicAdd(&deg[dst[e]], 1.0f);
}

__global__ void dinv_kernel(float* __restrict__ deg, int N) {
  int i = blockIdx.x * blockDim.x + threadIdx.x;
  if (i < N) deg[i] = rsqrtf(deg[i] + 1.0f);  // in place: deg -> dinv
}

// ---- layer 1 fused: t = x @ W1 (K=3); agg = t*dinv^2 + b1 (self-loop + bias) ----

__global__ void gemm1_fused_kernel(const float* __restrict__ x, const float* __restrict__ W1,
                                   const float* __restrict__ dinv, const float* __restrict__ b,
                                   float* __restrict__ t, float* __restrict__ agg, int N) {
  int idx = blockIdx.x * blockDim.x + threadIdx.x;
  if (idx >= N * FDIM) return;
  int n = idx >> 7;        // idx / 128
  int f = idx & (FDIM - 1);
  float x0 = x[n * 3 + 0], x1 = x[n * 3 + 1], x2 = x[n * 3 + 2];
  float v = fmaf(x0, W1[f], fmaf(x1, W1[FDIM + f], x2 * W1[2 * FDIM + f]));
  float di = dinv[n];
  t[idx] = v;
  agg[idx] = fmaf(v, di * di, b[f]);
}

// ---- layer 2 agg init (float4): agg = t*dinv^2 + b2 ----

__global__ void init_agg4_kernel(const float4* __restrict__ t4, const float* __restrict__ dinv,
                                 const float* __restrict__ b, float4* __restrict__ agg4, int N) {
  int idx = blockIdx.x * blockDim.x + threadIdx.x;   // over N * 32 float4s
  if (idx >= N * (FDIM / 4)) return;
  int n = idx >> 5;
  int f4 = (idx & 31) << 2;
  float di = dinv[n];
  float di2 = di * di;
  float4 v = t4[idx];
  float4 r;
  r.x = fmaf(v.x, di2, b[f4 + 0]);
  r.y = fmaf(v.y, di2, b[f4 + 1]);
  r.z = fmaf(v.z, di2, b[f4 + 2]);
  r.w = fmaf(v.w, di2, b[f4 + 3]);
  agg4[idx] = r;
}

// ---------------- edge scatter: one wave per edge, float4 per lane ----------------

__global__ void scatter_kernel(const float* __restrict__ t, float* __restrict__ agg,
                               const int* __restrict__ src, const int* __restrict__ dst,
                               const float* __restrict__ dinv, int E) {
  int gid = blockIdx.x * blockDim.x + threadIdx.x;
  int e = gid >> 5;
  if (e >= E) return;
  int l4 = (gid & 31) << 2;  // lane * 4
  int s = src[e], d = dst[e];
  float norm = dinv[s] * dinv[d];
  const float4 v = *(const float4*)(t + (size_t)s * FDIM + l4);
  float* base = agg + (size_t)d * FDIM + l4;
  atomicAdd(base + 0, v.x * norm);
  atomicAdd(base + 1, v.y * norm);
  atomicAdd(base + 2, v.z * norm);
  atomicAdd(base + 3, v.w * norm);
}

// ---------------- layer 2 GEMM via fp32 WMMA: C = relu(A) @ B ----
// A = agg1 (pre-relu hidden state, relu folded into load), B = W2 [128x128].
// One wave computes one 16x16 tile; K loop of 32 x V_WMMA_F32_16X16X4_F32.
// A (16x4 f32): lanes 0-15 hold K = k0+{0,1}; lanes 16-31 hold K = k0+{2,3}; M = lane&15.
// B (4x16):     mirrored, N = lane&15.
// C/D (16x16):  VGPR r -> row r + 8*(lane>>4), col = lane&15.
// Waves within a block share the same M tile (A reuse in WGP cache).

__global__ void gemm_wmma_kernel(const float* __restrict__ A, const float* __restrict__ B,
                                 float* __restrict__ C, int M) {
  int wave = (blockIdx.x * blockDim.x + threadIdx.x) >> 5;
  int lane = threadIdx.x & 31;
  const int nTilesN = FDIM / 16;               // 8 == waves per block
  int tm = wave / nTilesN;
  int tn = wave % nTilesN;
  if (tm * 16 >= M) return;                    // wave-uniform: EXEC stays all-ones
  int m = tm * 16 + (lane & 15);
  int n = tn * 16 + (lane & 15);
  int kh = (lane >> 4) * 2;                    // 0 or 2
  v8f c = {};
#pragma unroll
  for (int k0 = 0; k0 < FDIM; k0 += 4) {
    v2f a = *(const v2f*)(A + (size_t)m * FDIM + k0 + kh);
    a.x = fmaxf(a.x, 0.0f);                    // relu(h1) fused
    a.y = fmaxf(a.y, 0.0f);
    v2f b;
    b.x = B[(size_t)(k0 + kh) * FDIM + n];
    b.y = B[(size_t)(k0 + kh + 1) * FDIM + n];
    c = __builtin_amdgcn_wmma_f32_16x16x4_f32(false, a, false, b, (short)0, c,
                                              false, false);
  }
  int rowBase = tm * 16 + (lane >> 4) * 8;
#pragma unroll
  for (int r = 0; r < 8; ++r) {
    C[(size_t)(rowBase + r) * FDIM + n] = c[r];
  }
}

// ---------------- mean pool (relu fused) + final linear ----------------

__global__ void pool_kernel(const float* __restrict__ h, float* __restrict__ pooled, int N) {
  int f = threadIdx.x;  // 128 threads
  float s = 0.0f;
  for (int n = blockIdx.x; n < N; n += gridDim.x)
    s += fmaxf(h[(size_t)n * FDIM + f], 0.0f);       // relu(h2) fused
  atomicAdd(&pooled[f], s);
}

__global__ void final_kernel(const float* __restrict__ pooled, const float* __restrict__ Wf,
                             const float* __restrict__ bf, float* __restrict__ out, float invN) {
  __shared__ float sh[FDIM];
  int f = threadIdx.x;
  sh[f] = pooled[f] * invN * Wf[f];
  __syncthreads();
  for (int s = FDIM / 2; s > 0; s >>= 1) {
    if (f < s) sh[f] += sh[f + s];
    __syncthreads();
  }
  if (f == 0) out[0] = sh[0] + bf[0];
}

// ---------------- driver ----------------

extern "C" void kernel_launch(void* const* d_in, const int* in_sizes, int n_in,
                              void* d_out, int out_size, void* d_ws, size_t ws_size,
                              hipStream_t stream) {
  const float* x  = (const float*)d_in[0];
  const int*   ei = (const int*)d_in[1];
  const float* W1 = (const float*)d_in[2];
  const float* b1 = (const float*)d_in[3];
  const float* W2 = (const float*)d_in[4];
  const float* b2 = (const float*)d_in[5];
  const float* Wf = (const float*)d_in[6];
  const float* bf = (const float*)d_in[7];
  float* out = (float*)d_out;

  const int N = in_sizes[0] / 3;       // 100000
  const int E = in_sizes[1] / 2;       // 1600000
  const int* src = ei;
  const int* dst = ei + E;

  // workspace layout (floats)
  float* ws     = (float*)d_ws;
  float* dinv   = ws;                        // N  (deg, then rsqrt in place)
  float* t      = dinv + N;                  // N*FDIM  (pre-aggregation features)
  float* agg    = t + (size_t)N * FDIM;      // N*FDIM  (aggregated / hidden state)
  float* pooled = agg + (size_t)N * FDIM;    // FDIM

  const int B256 = 256;
  const int gNF   = (N * FDIM + B256 - 1) / B256;
  const int gNF4  = (N * (FDIM / 4) + B256 - 1) / B256;
  const int gN    = (N + B256 - 1) / B256;
  const int gE    = (E + B256 - 1) / B256;
  const int gE32  = (E * 32 + B256 - 1) / B256;     // one wave per edge
  const int nWave = (N / 16) * (FDIM / 16);         // WMMA tiles
  const int gWmma = (nWave * 32 + B256 - 1) / B256;

  // degrees -> dinv  (recomputed every call; deterministic)
  zero_kernel<<<gN, B256, 0, stream>>>(dinv, N);
  zero_kernel<<<1, B256, 0, stream>>>(pooled, FDIM);
  deg_kernel<<<gE, B256, 0, stream>>>(dst, dinv, E);
  dinv_kernel<<<gN, B256, 0, stream>>>(dinv, N);

  // ---- layer 1 ----
  gemm1_fused_kernel<<<gNF, B256, 0, stream>>>(x, W1, dinv, b1, t, agg, N);
  scatter_kernel<<<gE32, B256, 0, stream>>>(t, agg, src, dst, dinv, E);
  // agg = pre-relu h1; relu folded into WMMA A loads

  // ---- layer 2 ----
  gemm_wmma_kernel<<<gWmma, B256, 0, stream>>>(agg, W2, t, N);  // t = relu(h1) @ W2
  init_agg4_kernel<<<gNF4, B256, 0, stream>>>((const float4*)t, dinv, b2,
                                              (float4*)agg, N);
  scatter_kernel<<<gE32, B256, 0, stream>>>(t, agg, src, dst, dinv, E);
  // agg = pre-relu h2; relu folded into pooling

  // ---- mean pool + final linear ----
  pool_kernel<<<512, FDIM, 0, stream>>>(agg, pooled, N);
  final_kernel<<<1, FDIM, 0, stream>>>(pooled, Wf, bf, out, 1.0f / (float)N);
}